// GCN_66872640799057
// MI455X (gfx1250) — compile-verified
//
#include <hip/hip_runtime.h>
#include <hip/hip_bf16.h>
#include <math.h>

#define NN   50000
#define NE   800000
#define FIN  128
#define HIDN 128
#define NC   40
#define NCP  48
#define BN_EPS 1e-5f

typedef float v2f __attribute__((ext_vector_type(2)));
typedef float v8f __attribute__((ext_vector_type(8)));

__device__ __forceinline__ void atomAddF(float* p, float v) {
    unsafeAtomicAdd(p, v);   // native global_atomic_add_f32
}

// ---------------------------------------------------------------- utilities
__global__ void k_zero(unsigned int* p, int n) {
    int i = blockIdx.x * blockDim.x + threadIdx.x;
    if (i < n) p[i] = 0u;
}

__global__ void k_count_deg(const int* __restrict__ dst, unsigned int* __restrict__ deg) {
    int i = blockIdx.x * blockDim.x + threadIdx.x;
    if (i < NE) atomicAdd(&deg[dst[i]], 1u);
}

__global__ void k_dinv(const unsigned int* __restrict__ deg, float* __restrict__ dinv) {
    int i = blockIdx.x * blockDim.x + threadIdx.x;
    if (i < NN) {
        float d = (float)(deg[i] + 1u);  // + self loop, always >= 1
        dinv[i] = rsqrtf(d);
    }
}

// ---------------------------------------------------------------- fp32 WMMA GEMM
// C[NN x ldc] = A[NN x K] * B[K x nb (stride ldb)]   (cols >= nb zero-masked)
// One 16x16 tile per wave via V_WMMA_F32_16X16X4_F32, K-loop step 4.
// A layout (16x4 f32): lane l<16 -> {K=0,1}, lanes 16-31 -> {K=2,3} (contiguous pair)
// B layout (4x16 f32): v0 = B[2*half][col], v1 = B[2*half+1][col]
__global__ void k_gemm_wmma(const float* __restrict__ A, const float* __restrict__ B,
                            float* __restrict__ C, int K, int nb, int ldb, int ldc) {
    const int row0 = blockIdx.x << 4;
    const int wave = threadIdx.x >> 5;
    const int lane = threadIdx.x & 31;
    const int half = lane >> 4;
    const int l    = lane & 15;
    const int col0 = wave << 4;
    const int col  = col0 + l;
    const int   bcol  = (col < nb) ? col : 0;
    const float bmask = (col < nb) ? 1.0f : 0.0f;

    const float* ap = A + (size_t)(row0 + l) * K + (half << 1);
    const float* bp = B + (size_t)(half << 1) * ldb + bcol;

    v8f acc = {};
#pragma unroll 8
    for (int k0 = 0; k0 < K; k0 += 4) {
        v2f a = *(const v2f*)(ap + k0);               // 8B aligned (k0, half*2 even)
        v2f b;
        b.x = bp[(size_t)k0 * ldb] * bmask;
        b.y = bp[(size_t)(k0 + 1) * ldb] * bmask;
        acc = __builtin_amdgcn_wmma_f32_16x16x4_f32(
            /*neg_a=*/false, a, /*neg_b=*/false, b,
            /*c_mod=*/(short)0, acc, /*reuse_a=*/false, /*reuse_b=*/false);
    }
    float* cp = C + (size_t)(row0 + (half << 3)) * ldc + col0 + l;
#pragma unroll
    for (int r = 0; r < 8; ++r) cp[(size_t)r * ldc] = acc[r];
}

// ---------------------------------------------------------------- aggregation
// out[i,f] = H[i,f]*dinv[i]^2 + bias[f]   (self-loop term + bias)
__global__ void k_agg_init(const float* __restrict__ H, const float* __restrict__ dinv,
                           const float* __restrict__ bias, float* __restrict__ out,
                           int F, int ldh, int ldo) {
    int i = blockIdx.x;
    int f = threadIdx.x;
    if (f >= F) return;
    float di = dinv[i];
    out[(size_t)i * ldo + f] = H[(size_t)i * ldh + f] * di * di + bias[f];
}

// out[dst] += H[src] * dinv[src]*dinv[dst]   (one wave per edge)
__global__ void k_agg_edges(const int* __restrict__ src, const int* __restrict__ dst,
                            const float* __restrict__ dinv, const float* __restrict__ H,
                            float* __restrict__ out, int F, int ldh, int ldo) {
    int e    = blockIdx.x * (blockDim.x >> 5) + (threadIdx.x >> 5);
    int lane = threadIdx.x & 31;
    if (e >= NE) return;
    int s = src[e], d = dst[e];
    float nrm = dinv[s] * dinv[d];
    const float* hp = H + (size_t)s * ldh;
    float*       op = out + (size_t)d * ldo;
    for (int f = lane; f < F; f += 32)
        atomAddF(&op[f], hp[f] * nrm);
}

// ---------------------------------------------------------------- batchnorm
__global__ void k_bn_stats(const float* __restrict__ X, float* __restrict__ stats) {
    int c = threadIdx.x;                         // 128 threads = 128 columns
    float s = 0.0f, s2 = 0.0f;
    for (int r = blockIdx.x; r < NN; r += gridDim.x) {
        float v = X[(size_t)r * HIDN + c];
        s += v; s2 += v * v;
    }
    atomAddF(&stats[c], s);
    atomAddF(&stats[HIDN + c], s2);
}

__global__ void k_bn_final(float* __restrict__ stats) {
    int c = threadIdx.x;
    float mean = stats[c] * (1.0f / (float)NN);
    float var  = stats[HIDN + c] * (1.0f / (float)NN) - mean * mean;
    stats[2 * HIDN + c] = mean;
    stats[3 * HIDN + c] = rsqrtf(var + BN_EPS);
}

__global__ void k_bn_apply(const float* __restrict__ X, const float* __restrict__ stats,
                           const float* __restrict__ gamma, const float* __restrict__ beta,
                           float* __restrict__ Y) {
    int idx = blockIdx.x * blockDim.x + threadIdx.x;
    if (idx >= NN * HIDN) return;
    int c = idx & (HIDN - 1);
    float v = (X[idx] - stats[2 * HIDN + c]) * stats[3 * HIDN + c] * gamma[c] + beta[c];
    Y[idx] = fmaxf(v, 0.0f);
}

// ---------------------------------------------------------------- log_softmax (in place, 40 cols)
__global__ void k_logsoftmax(float* __restrict__ out) {
    int wave = threadIdx.x >> 5;
    int lane = threadIdx.x & 31;
    int row  = blockIdx.x * 8 + wave;
    if (row >= NN) return;
    float* p = out + (size_t)row * NC;
    float v0 = p[lane];
    float v1 = (lane < NC - 32) ? p[32 + lane] : -INFINITY;
    float m = fmaxf(v0, v1);
    for (int off = 16; off; off >>= 1) m = fmaxf(m, __shfl_xor(m, off, 32));
    float s = expf(v0 - m) + ((lane < NC - 32) ? expf(v1 - m) : 0.0f);
    for (int off = 16; off; off >>= 1) s += __shfl_xor(s, off, 32);
    float lse = m + logf(s);
    p[lane] = v0 - lse;
    if (lane < NC - 32) p[32 + lane] = v1 - lse;
}

// ---------------------------------------------------------------- driver
extern "C" void kernel_launch(void* const* d_in, const int* in_sizes, int n_in,
                              void* d_out, int out_size, void* d_ws, size_t ws_size,
                              hipStream_t stream) {
    const float* feat = (const float*)d_in[0];
    const int*   ei   = (const int*)d_in[1];
    const int*   srcp = ei;
    const int*   dstp = ei + NE;
    const float* W1 = (const float*)d_in[2];
    const float* b1 = (const float*)d_in[3];
    const float* g1 = (const float*)d_in[4];
    const float* be1 = (const float*)d_in[5];
    const float* W2 = (const float*)d_in[6];
    const float* b2 = (const float*)d_in[7];
    const float* g2 = (const float*)d_in[8];
    const float* be2 = (const float*)d_in[9];
    const float* W3 = (const float*)d_in[10];
    const float* b3 = (const float*)d_in[11];
    float* out = (float*)d_out;

    char* ws = (char*)d_ws;
    unsigned int* deg = (unsigned int*)ws;  ws += (size_t)NN * sizeof(unsigned int);
    float* dinv = (float*)ws;               ws += (size_t)NN * sizeof(float);
    float* buf1 = (float*)ws;               ws += (size_t)NN * HIDN * sizeof(float);
    float* buf2 = (float*)ws;               ws += (size_t)NN * HIDN * sizeof(float);
    float* stats = (float*)ws;              ws += 4 * HIDN * sizeof(float);

    // normalization coefficients
    k_zero<<<(NN + 255) / 256, 256, 0, stream>>>(deg, NN);
    k_count_deg<<<(NE + 255) / 256, 256, 0, stream>>>(dstp, deg);
    k_dinv<<<(NN + 255) / 256, 256, 0, stream>>>(deg, dinv);

    // ---- layer 1: GEMM -> aggregate -> BN+ReLU
    k_gemm_wmma<<<NN / 16, 256, 0, stream>>>(feat, W1, buf1, FIN, HIDN, HIDN, HIDN);
    k_agg_init<<<NN, HIDN, 0, stream>>>(buf1, dinv, b1, buf2, HIDN, HIDN, HIDN);
    k_agg_edges<<<(NE + 7) / 8, 256, 0, stream>>>(srcp, dstp, dinv, buf1, buf2, HIDN, HIDN, HIDN);
    k_zero<<<1, 256, 0, stream>>>((unsigned int*)stats, 2 * HIDN);
    k_bn_stats<<<512, HIDN, 0, stream>>>(buf2, stats);
    k_bn_final<<<1, HIDN, 0, stream>>>(stats);
    k_bn_apply<<<(NN * HIDN + 255) / 256, 256, 0, stream>>>(buf2, stats, g1, be1, buf1);

    // ---- layer 2
    k_gemm_wmma<<<NN / 16, 256, 0, stream>>>(buf1, W2, buf2, HIDN, HIDN, HIDN, HIDN);
    k_agg_init<<<NN, HIDN, 0, stream>>>(buf2, dinv, b2, buf1, HIDN, HIDN, HIDN);
    k_agg_edges<<<(NE + 7) / 8, 256, 0, stream>>>(srcp, dstp, dinv, buf2, buf1, HIDN, HIDN, HIDN);
    k_zero<<<1, 256, 0, stream>>>((unsigned int*)stats, 2 * HIDN);
    k_bn_stats<<<512, HIDN, 0, stream>>>(buf1, stats);
    k_bn_final<<<1, HIDN, 0, stream>>>(stats);
    k_bn_apply<<<(NN * HIDN + 255) / 256, 256, 0, stream>>>(buf1, stats, g2, be2, buf2);

    // ---- layer 3: GEMM (40 cols padded to 48) -> aggregate into d_out -> log_softmax
    float* H3 = buf1;  // NN x 48 fits in NN x 128
    k_gemm_wmma<<<NN / 16, 96, 0, stream>>>(buf2, W3, H3, HIDN, NC, NC, NCP);
    k_agg_init<<<NN, 64, 0, stream>>>(H3, dinv, b3, out, NC, NCP, NC);
    k_agg_edges<<<(NE + 7) / 8, 256, 0, stream>>>(srcp, dstp, dinv, H3, out, NC, NCP, NC);
    k_logsoftmax<<<NN / 8, 256, 0, stream>>>(out);
}